// INFLOODClassifier_40149354283398
// MI455X (gfx1250) — compile-verified
//
#include <hip/hip_runtime.h>

typedef float v2f __attribute__((ext_vector_type(2)));
typedef float v8f __attribute__((ext_vector_type(8)));
typedef unsigned int u32x4 __attribute__((ext_vector_type(4)));
typedef int i32x4 __attribute__((ext_vector_type(4)));
typedef int i32x8 __attribute__((ext_vector_type(8)));

#define NN    8192
#define DD    128
#define KK    20
#define CC    10
#define NCH   4          // column chunks
#define CHW   (NN/NCH)   // 2048 columns per chunk
#define BS_STRIDE 132    // padded LDS row stride (floats) -> conflict-free b64 reads
#define TILE_BYTES (16 * BS_STRIDE * 4)
#define FLT_BIG 3.4e38f

// ---------------------------------------------------------------------------
// TDM: DMA one 16-row x 128-float tile (row-major) global -> LDS, with the
// hardware inserting 4 DWORDs of padding every 128 DWORDs so the LDS image
// has row stride 132 floats (matches the WMMA B-fragment fetch pattern).
// Descriptor per CDNA5 ISA ch.8 (D# groups); this toolchain exposes the
// 6-arg builtin: (u32x4 g0, i32x8 g1, i32x4, i32x4, i32x8, i32 cpol).
// ---------------------------------------------------------------------------
__device__ __forceinline__ void tdm_load_tile(const float* gsrc, unsigned lds_addr) {
    unsigned long long ga = (unsigned long long)(const void*)gsrc;
    u32x4 g0 = {
        1u,                                   // [1:0] count=1 (valid), rest 0
        lds_addr,                             // [63:32] LDS byte address
        (unsigned)ga,                         // [95:64] global addr lo
        ((unsigned)(ga >> 32) & 0x01FFFFFFu)  // [120:96] global addr hi
            | 0x80000000u                     // [127:126] type = 2 ("image")
    };
    i32x8 g1 = {
        (int)0x07920000,      // data_size=4B([17:16]=2), pad_enable(bit20),
                              // pad_interval=128DW([24:22]=6), pad_amount=4DW([31:25]=3)
        (int)(128u << 16),    // tensor_dim0 = 128   (bits 79:48, low half)
        (int)(8192u << 16),   // tensor_dim1 = 8192  (bits 111:80, low half)
        (int)(128u << 16),    // tile_dim0  = 128    (bits 127:112)
        16,                   // tile_dim1  = 16     (bits 143:128)
        128,                  // tensor_dim0_stride = 128 (bits 207:160, low 32)
        0, 0
    };
    i32x4 z4 = {0, 0, 0, 0};             // groups 2/3 unused for 2-D tiles
    i32x8 z8 = {0, 0, 0, 0, 0, 0, 0, 0}; // trailing group (unused, zero)
    __builtin_amdgcn_tensor_load_to_lds(g0, g1, z4, z4, z8, 0);
}

// ---------------------------------------------------------------------------
// Kernel 1: row squared norms
// ---------------------------------------------------------------------------
__global__ __launch_bounds__(256) void k_sqnorm(const float* __restrict__ X,
                                                float* __restrict__ sq) {
    int row = blockIdx.x * blockDim.x + threadIdx.x;
    const float4* p = (const float4*)(X + (size_t)row * DD);
    float s = 0.f;
#pragma unroll 8
    for (int i = 0; i < DD / 4; i++) {
        float4 v = p[i];
        s += v.x * v.x + v.y * v.y + v.z * v.z + v.w * v.w;
    }
    sq[row] = s;
}

// ---------------------------------------------------------------------------
// Kernel 2: fused f32-WMMA Gram tile + per-row partial top-K over one column
// chunk. Grid: 256 blocks (64 row-blocks x 4 chunks), 256 threads (8 waves).
// B tiles are staged by the Tensor Data Mover, double-buffered: wave 0 issues
// the DMA for tile t+1 while all waves compute tile t.
// ---------------------------------------------------------------------------
__global__ __launch_bounds__(256) void k_topk(const float* __restrict__ X,
                                              const float* __restrict__ sq,
                                              float* __restrict__ pdist,
                                              int* __restrict__ pidx) {
    __shared__ float Bs[2][16 * BS_STRIDE];    // TDM-staged column tiles
    __shared__ float tl[8][256];               // per-wave 16x16 d2 tile
    __shared__ float topd[8][16][KK];          // per-row top-K dists (squared)
    __shared__ int   topi[8][16][KK];          // per-row top-K indices

    const int tid  = threadIdx.x;
    const int wave = tid >> 5;
    const int lane = tid & 31;
    const int lm   = lane & 15;
    const int hi   = lane >> 4;

    const int bx      = blockIdx.x;
    const int chunk   = bx & (NCH - 1);
    const int rowblk  = bx >> 2;
    const int wrow    = rowblk * 128 + wave * 16;  // wave's first row
    const int colbase = chunk * CHW;

    // LDS byte address of the staging buffers (generic LDS ptr low 32 bits
    // are the LDS byte offset per CDNA5 flat-aperture rules)
    const unsigned ldsBs = (unsigned)(size_t)(void*)&Bs[0][0];

    // init per-row top-K state
    float thr = FLT_BIG;
    int maxpos = 0;
    if (lane < 16) {
#pragma unroll
        for (int k = 0; k < KK; k++) {
            topd[wave][lane][k] = FLT_BIG;
            topi[wave][lane][k] = 0;
        }
    }

    // Preload A fragments (16x128 f32) in WMMA A layout for 16x16x4_f32:
    // lanes 0-15: M=lane, K={4kk,4kk+1}; lanes 16-31: M=lane-16, K={4kk+2,4kk+3}
    v2f a[32];
    {
        const float* arow = X + (size_t)(wrow + lm) * DD + hi * 2;
#pragma unroll
        for (int kk = 0; kk < 32; kk++) {
            a[kk].x = arow[kk * 4 + 0];
            a[kk].y = arow[kk * 4 + 1];
        }
    }
    // per-lane row norms for the 8 accumulator rows (C layout: m = r + 8*hi)
    float sqi[8];
#pragma unroll
    for (int r = 0; r < 8; r++) sqi[r] = sq[wrow + r + 8 * hi];

    // prologue: DMA tile 0 into buffer 0
    if (wave == 0) tdm_load_tile(X + (size_t)colbase * DD, ldsBs);

    for (int t = 0; t < CHW / 16; t++) {
        const int c0  = colbase + t * 16;
        const int cur = t & 1;
        if (wave == 0) __builtin_amdgcn_s_wait_tensorcnt(0);  // tile t landed
        __syncthreads();   // tile t visible to all waves; tile t-1 readers done
        if (wave == 0 && t + 1 < CHW / 16)                    // prefetch t+1
            tdm_load_tile(X + (size_t)(c0 + 16) * DD,
                          ldsBs + (unsigned)((t + 1) & 1) * TILE_BYTES);

        v8f c = {0.f, 0.f, 0.f, 0.f, 0.f, 0.f, 0.f, 0.f};
#pragma unroll
        for (int kk = 0; kk < 32; kk++) {
            v2f b = *(const v2f*)&Bs[cur][lm * BS_STRIDE + kk * 4 + hi * 2];
            c = __builtin_amdgcn_wmma_f32_16x16x4_f32(
                    false, a[kk], false, b, (short)0, c, false, false);
        }

        // epilogue: d2 = |xi|^2 + |xj|^2 - 2 xi.xj -> per-wave LDS tile
        float sqj = sq[c0 + lm];   // L2-resident
#pragma unroll
        for (int r = 0; r < 8; r++) {
            float d2 = sqi[r] + sqj - 2.0f * c[r];
            tl[wave][(r + 8 * hi) * 16 + lm] = d2;
        }

        // owner-lane top-K maintenance (threshold filter; inserts are rare)
        if (lane < 16) {
            const int m = lane;
#pragma unroll 4
            for (int n = 0; n < 16; n++) {
                float d = tl[wave][m * 16 + n];
                if (d < thr) {
                    topd[wave][m][maxpos] = d;
                    topi[wave][m][maxpos] = c0 + n;
                    float mx = -FLT_BIG; int mp = 0;
#pragma unroll
                    for (int k = 0; k < KK; k++) {
                        float v = topd[wave][m][k];
                        if (v > mx) { mx = v; mp = k; }
                    }
                    thr = mx; maxpos = mp;
                }
            }
        }
    }

    // write partial top-K for this (row, chunk)
    if (lane < 16) {
        int row = wrow + lane;
#pragma unroll
        for (int k = 0; k < KK; k++) {
            pdist[((size_t)row * NCH + chunk) * KK + k] = topd[wave][lane][k];
            pidx [((size_t)row * NCH + chunk) * KK + k] = topi[wave][lane][k];
        }
    }
}

// ---------------------------------------------------------------------------
// Kernel 2b: merge 4 partial top-20 lists (80 candidates) -> final top-20,
// density = 1/(mean sqrt(d2) + eps), and final knn index list.
// ---------------------------------------------------------------------------
__global__ __launch_bounds__(256) void k_merge(const float* __restrict__ pdist,
                                               const int* __restrict__ pidx,
                                               float* __restrict__ density,
                                               int* __restrict__ knn) {
    __shared__ float md[256][KK];
    __shared__ int   mi[256][KK];
    int tid = threadIdx.x;
    int row = blockIdx.x * 256 + tid;

    float thr = FLT_BIG; int maxpos = 0;
#pragma unroll
    for (int k = 0; k < KK; k++) { md[tid][k] = FLT_BIG; mi[tid][k] = 0; }

    for (int cidx = 0; cidx < NCH * KK; cidx++) {
        float d = pdist[(size_t)row * (NCH * KK) + cidx];
        int   j = pidx [(size_t)row * (NCH * KK) + cidx];
        if (d < thr) {
            md[tid][maxpos] = d; mi[tid][maxpos] = j;
            float mx = -FLT_BIG; int mp = 0;
#pragma unroll
            for (int k = 0; k < KK; k++) {
                if (md[tid][k] > mx) { mx = md[tid][k]; mp = k; }
            }
            thr = mx; maxpos = mp;
        }
    }
    float s = 0.f;
#pragma unroll
    for (int k = 0; k < KK; k++) {
        s += sqrtf(fmaxf(md[tid][k], 0.f));
        knn[(size_t)row * KK + k] = mi[tid][k];
    }
    density[row] = 1.0f / (s * (1.0f / KK) + 1e-10f);
}

// ---------------------------------------------------------------------------
// Kernel 3: zero the N x N bitset (N/32 words per row)
// ---------------------------------------------------------------------------
__global__ __launch_bounds__(256) void k_zero(unsigned* __restrict__ p, int n) {
    int i = blockIdx.x * blockDim.x + threadIdx.x;
    if (i < n) p[i] = 0u;
}

// ---------------------------------------------------------------------------
// Kernel 3b: symmetric scatter -> mask[i][j] |= 1, mask[j][i] |= 1
// ---------------------------------------------------------------------------
__global__ __launch_bounds__(256) void k_scatter(const int* __restrict__ knn,
                                                 unsigned* __restrict__ mask) {
    int t = blockIdx.x * blockDim.x + threadIdx.x;   // 0 .. N*K-1
    int i = t / KK;
    int j = knn[t];
    atomicOr(&mask[(size_t)i * (NN / 32) + (j >> 5)], 1u << (j & 31));
    atomicOr(&mask[(size_t)j * (NN / 32) + (i >> 5)], 1u << (i & 31));
}

// ---------------------------------------------------------------------------
// Kernel 4: one wave per row. Gather densities over set mask bits, wave32
// reduce, score, OOD flag, argmax over logits. Outputs concatenated:
// [0,N) scores, [N,2N) flags, [2N,3N) class preds.
// ---------------------------------------------------------------------------
__global__ __launch_bounds__(256) void k_score(const unsigned* __restrict__ mask,
                                               const float* __restrict__ density,
                                               const float* __restrict__ logits,
                                               float* __restrict__ out) {
    int wave = threadIdx.x >> 5;
    int lane = threadIdx.x & 31;
    int row  = blockIdx.x * 8 + wave;

    float s = 0.f;
    int   cnt = 0;
#pragma unroll
    for (int w = 0; w < (NN / 32) / 32; w++) {       // 8 words per lane
        int widx = w * 32 + lane;
        unsigned word = mask[(size_t)row * (NN / 32) + widx];
        while (word) {
            int b = __builtin_ctz(word);
            word &= word - 1;
            s += density[widx * 32 + b];
            cnt++;
        }
    }
    // wave32 reduction
#pragma unroll
    for (int off = 16; off > 0; off >>= 1) {
        s   += __shfl_xor(s, off, 32);
        cnt += __shfl_xor(cnt, off, 32);
    }
    if (lane == 0) {
        float dens  = density[row];
        float avg   = s / fmaxf((float)cnt, 1.0f);
        float score = -(dens / (avg + 1e-10f));
        bool ood    = score < -0.5f;
        int pred;
        if (ood) {
            pred = -1;
        } else {
            float bm = -FLT_BIG; int bi = 0;
#pragma unroll
            for (int cl = 0; cl < CC; cl++) {
                float v = logits[(size_t)row * CC + cl];
                if (v > bm) { bm = v; bi = cl; }
            }
            pred = bi;
        }
        out[row]          = score;
        out[NN + row]     = ood ? 1.0f : 0.0f;
        out[2 * NN + row] = (float)pred;
    }
}

// ---------------------------------------------------------------------------
extern "C" void kernel_launch(void* const* d_in, const int* in_sizes, int n_in,
                              void* d_out, int out_size, void* d_ws, size_t ws_size,
                              hipStream_t stream) {
    const float* X      = (const float*)d_in[0];   // [N, D] f32
    const float* logits = (const float*)d_in[1];   // [N, C] f32
    float* out = (float*)d_out;

    // workspace layout (bytes)
    char* ws = (char*)d_ws;
    float*    sq      = (float*)(ws);                                  // N
    float*    density = (float*)(ws + 32768);                          // N
    float*    pdist   = (float*)(ws + 65536);                          // N*4*20
    int*      pidx    = (int*)  (ws + 65536 + 2621440);                // N*4*20
    int*      knn     = (int*)  (ws + 65536 + 2 * 2621440);            // N*20
    unsigned* mask    = (unsigned*)(ws + 65536 + 2 * 2621440 + 655360);// N*N/32

    // 1) row norms
    k_sqnorm<<<NN / 256, 256, 0, stream>>>(X, sq);

    // 3) zero bitset (independent of 2; issue early)
    int nwords = NN * (NN / 32);
    k_zero<<<(nwords + 255) / 256, 256, 0, stream>>>(mask, nwords);

    // 2) fused WMMA GEMM + partial top-K  (64 row-blocks x 4 column chunks)
    k_topk<<<(NN / 128) * NCH, 256, 0, stream>>>(X, sq, pdist, pidx);

    // 2b) merge partials -> density + knn
    k_merge<<<NN / 256, 256, 0, stream>>>(pdist, pidx, density, knn);

    // 3b) symmetric kNN-union bitset
    k_scatter<<<(NN * KK) / 256, 256, 0, stream>>>(knn, mask);

    // 4) scores / flags / preds
    k_score<<<NN / 8, 256, 0, stream>>>(mask, density, logits, out);
}